// GraphConvolution_10642928959666
// MI455X (gfx1250) — compile-verified
//
#include <hip/hip_runtime.h>
#include <hip/hip_bf16.h>

// ---------------------------------------------------------------------------
// GraphConvolution: out = segment_sum(edge_val * x[edge_dst], edge_src) @ W + b
//   N = 100000 nodes, E = 1.6M edges, D_in = D_out = 128, all float32.
//
// Phase 0: zero the aggregation buffer (d_ws, N*128 floats = 51.2 MB).
// Phase 1: per-edge scatter. One wave32 per edge: coalesced float4 row gather
//          of x[dst] (512 B), scale by edge_val, 4x global_atomic_add_f32 per
//          lane into agg[src]. x and agg both fit in the 192 MB L2.
// Phase 2: dense GEMM agg @ W + b via V_WMMA_F32_16X16X4_F32 (f32 A/B/C keeps
//          full reference precision). One wave per 16x16 tile, K-loop of 32
//          chunks of K=4.
// ---------------------------------------------------------------------------

typedef __attribute__((ext_vector_type(2))) float v2f;
typedef __attribute__((ext_vector_type(8))) float v8f;

#define D 128            // feature width (both in and out)
#define WAVES_PER_BLOCK 8

__global__ void gc_zero_kernel(float4* __restrict__ agg, int n4) {
    int i = blockIdx.x * blockDim.x + threadIdx.x;
    if (i < n4) {
        agg[i] = make_float4(0.f, 0.f, 0.f, 0.f);
    }
}

// One wave (32 lanes) per edge. Lane L handles floats [4L, 4L+4) of the row.
__global__ void gc_scatter_kernel(const float* __restrict__ x,
                                  const int* __restrict__ edge_src,
                                  const int* __restrict__ edge_dst,
                                  const float* __restrict__ edge_val,
                                  float* __restrict__ agg,
                                  int n_edges) {
    const int wave = threadIdx.x >> 5;
    const int lane = threadIdx.x & 31;
    const int e = blockIdx.x * WAVES_PER_BLOCK + wave;
    if (e >= n_edges) return;

    const int   src = edge_src[e];
    const int   dst = edge_dst[e];
    const float val = edge_val[e];

    // Coalesced 512B row read: lane L -> float4 at column 4L.
    const float4 v = ((const float4*)(x + (size_t)dst * D))[lane];

    float* a = agg + (size_t)src * D + lane * 4;
    atomicAdd(a + 0, val * v.x);
    atomicAdd(a + 1, val * v.y);
    atomicAdd(a + 2, val * v.z);
    atomicAdd(a + 3, val * v.w);
}

// GEMM: out[N x 128] = agg[N x 128] @ W[128 x 128] + b[128]
// Grid: one block per 16-row tile; 8 waves cover the 8 column tiles of 16.
// V_WMMA_F32_16X16X4_F32 fragment layouts (ISA 7.12.2):
//   A 16x4 : lane l holds A[l&15][2*(l>>4)+0 .. +1]  (2 VGPRs)
//   B 4x16 : lane l holds B[2*(l>>4)+0 .. +1][l&15]  (2 VGPRs)
//   C 16x16: VGPR r, lane l -> row r + 8*(l>>4), col l&15 (8 VGPRs)
__global__ void gc_gemm_kernel(const float* __restrict__ agg,
                               const float* __restrict__ W,
                               const float* __restrict__ bias,
                               float* __restrict__ out,
                               int n_nodes) {
    const int wave = threadIdx.x >> 5;     // 0..7 -> column tile
    const int lane = threadIdx.x & 31;

    const int row_base = blockIdx.x * 16;
    const int col_base = wave * 16;

    const int m16  = lane & 15;            // row (A) / col (B,C) within tile
    const int half = lane >> 4;            // 0 or 1: K sub-pair / row-half

    // Clamp A-row for ragged final tile (stores are guarded below).
    int a_row = row_base + m16;
    if (a_row > n_nodes - 1) a_row = n_nodes - 1;
    const float* a_base = agg + (size_t)a_row * D + half * 2;

    const int b_col = col_base + m16;

    v8f acc = {0.f, 0.f, 0.f, 0.f, 0.f, 0.f, 0.f, 0.f};

    #pragma unroll
    for (int kb = 0; kb < D; kb += 4) {
        // A fragment: 2 consecutive f32 (8B aligned since kb + 2*half is even)
        v2f afrag = *(const v2f*)(a_base + kb);
        // B fragment: two rows of W at this column
        v2f bfrag;
        bfrag.x = W[(size_t)(kb + half * 2 + 0) * D + b_col];
        bfrag.y = W[(size_t)(kb + half * 2 + 1) * D + b_col];
        // 8-arg form: (neg_a, A, neg_b, B, c_mod, C, reuse_a, reuse_b)
        acc = __builtin_amdgcn_wmma_f32_16x16x4_f32(
            false, afrag, false, bfrag, (short)0, acc, false, false);
    }

    const float bv = bias[b_col];
    #pragma unroll
    for (int r = 0; r < 8; ++r) {
        const int out_row = row_base + half * 8 + r;
        if (out_row < n_nodes) {
            out[(size_t)out_row * D + b_col] = acc[r] + bv;
        }
    }
}

extern "C" void kernel_launch(void* const* d_in, const int* in_sizes, int n_in,
                              void* d_out, int out_size, void* d_ws, size_t ws_size,
                              hipStream_t stream) {
    const float* x        = (const float*)d_in[0];
    const int*   edge_src = (const int*)d_in[1];
    const int*   edge_dst = (const int*)d_in[2];
    const float* edge_val = (const float*)d_in[3];
    const float* W        = (const float*)d_in[4];
    const float* bias     = (const float*)d_in[5];
    float*       out      = (float*)d_out;
    float*       agg      = (float*)d_ws;   // N*128 floats of scratch

    const int n_nodes = in_sizes[0] / D;
    const int n_edges = in_sizes[1];

    // Phase 0: zero agg
    {
        const int n4 = n_nodes * (D / 4);
        const int threads = 256;
        const int blocks = (n4 + threads - 1) / threads;
        gc_zero_kernel<<<blocks, threads, 0, stream>>>((float4*)agg, n4);
    }

    // Phase 1: edge scatter (one wave32 per edge)
    {
        const int threads = WAVES_PER_BLOCK * 32;  // 256
        const int blocks = (n_edges + WAVES_PER_BLOCK - 1) / WAVES_PER_BLOCK;
        gc_scatter_kernel<<<blocks, threads, 0, stream>>>(
            x, edge_src, edge_dst, edge_val, agg, n_edges);
    }

    // Phase 2: WMMA GEMM + bias
    {
        const int threads = WAVES_PER_BLOCK * 32;  // 256 = 8 waves
        const int blocks = (n_nodes + 15) / 16;    // one 16-row tile per block
        gc_gemm_kernel<<<blocks, threads, 0, stream>>>(
            agg, W, bias, out, n_nodes);
    }
}